// Prediction_78700980732158
// MI455X (gfx1250) — compile-verified
//
#include <hip/hip_runtime.h>
#include <stdint.h>

// ---------------------------------------------------------------------------
// CenterNet-style decode for MI455X (gfx1250, wave32).
// Memory-bound stencil + exact thresholded top-k. No GEMM -> no WMMA; the
// CDNA5 feature used is GLOBAL_LOAD_ASYNC_TO_LDS_B128 (+ s_wait_asynccnt)
// to stage heatmap tiles into LDS for the 3x3 NMS stencil.
// Workspace requirement: ~787 KB.
// ---------------------------------------------------------------------------

namespace {
constexpr int kB = 16, kC = 80, kH = 256, kW = 256;
constexpr int kHW = kH * kW;
constexpr int kTopK = 100;
constexpr float kThresh = 0.01f;
constexpr float kScale = 4.0f;
constexpr int kBins = 4096;
constexpr int kCandCap = 4096;

// 32-bit word offsets inside d_ws
constexpr int WS_HIST   = 0;                          // kB * kBins counters
constexpr int WS_CUT    = WS_HIST + kB * kBins;       // kB cutoff bins
constexpr int WS_CNT    = WS_CUT + kB;                // kB candidate counts
constexpr int WS_CAND_S = WS_CNT + kB;                // kB * kCandCap floats
constexpr int WS_CAND_I = WS_CAND_S + kB * kCandCap;  // kB * kCandCap u32
constexpr int WS_ZERO_WORDS = WS_CAND_S;              // hist+cut+cnt zeroed per call
}  // namespace

#if __has_builtin(__builtin_amdgcn_global_load_async_to_lds_b128)
#define ATHENA_HAVE_ASYNC_LDS 1
#endif

typedef int athena_v4i __attribute__((ext_vector_type(4)));
typedef __attribute__((address_space(1))) athena_v4i* athena_gptr_v4i;
typedef __attribute__((address_space(3))) athena_v4i* athena_lptr_v4i;

__device__ __forceinline__ void stage16(const float* gp, float* lp) {
#ifdef ATHENA_HAVE_ASYNC_LDS
  // CDNA5 async DMA: global -> LDS, 16B per lane, tracked by ASYNCcnt.
  __builtin_amdgcn_global_load_async_to_lds_b128(
      (athena_gptr_v4i)gp,
      (athena_lptr_v4i)lp,
      /*imm offset*/ 0, /*cpol*/ 0);
#else
  *(float4*)lp = *(const float4*)gp;
#endif
}

__device__ __forceinline__ void stage_wait() {
#ifdef ATHENA_HAVE_ASYNC_LDS
#if __has_builtin(__builtin_amdgcn_s_wait_asynccnt)
  __builtin_amdgcn_s_wait_asynccnt(0);
#else
  asm volatile("s_wait_asynccnt 0" ::: "memory");
#endif
#endif
}

// ---------------------------------------------------------------------------
// Kernel 0: zero histogram / cutoff / counters.
// ---------------------------------------------------------------------------
__global__ void init_ws_kernel(unsigned* __restrict__ ws) {
  int i = blockIdx.x * blockDim.x + threadIdx.x;
  if (i < WS_ZERO_WORDS) ws[i] = 0u;
}

// ---------------------------------------------------------------------------
// Kernels 1 & 3: 3x3 NMS stencil over LDS-staged tiles.
//   COLLECT=false: histogram peak scores (> thresh) per batch.
//   COLLECT=true : append (score, index) for peaks whose bin >= cutoff[b].
// Block = 256 threads handles (b, c, 4 rows). Halo rows edge-clamped, which
// is exactly equivalent to the SAME-padded max window.
// ---------------------------------------------------------------------------
template <bool COLLECT>
__global__ void peaks_kernel(const float* __restrict__ hm,
                             unsigned* __restrict__ ws) {
  __shared__ float tile[6 * kW];  // rows y0-1 .. y0+4

  const int t = threadIdx.x;
  const int rowBlocks = kH / 4;                 // 64
  const int bc = blockIdx.x / rowBlocks;        // b*kC + c
  const int rowblk = blockIdx.x % rowBlocks;
  const int b = bc / kC;
  const int c = bc % kC;
  const int y0 = rowblk * 4;
  const float* gbase = hm + (size_t)bc * kHW;

  // Stage 6 rows x 256 floats = 384 chunks of 16B into LDS.
  for (int ch = t; ch < 384; ch += 256) {
    int lrow = ch >> 4;                         // 16 chunks per row
    int gr = y0 - 1 + lrow;
    gr = gr < 0 ? 0 : (gr > kH - 1 ? kH - 1 : gr);
    const float* gp = gbase + gr * kW + ((ch & 15) << 2);
    stage16(gp, &tile[ch << 2]);
  }
  stage_wait();
  __syncthreads();

  const unsigned cut = COLLECT ? ws[WS_CUT + b] : 0u;

  const int r = t >> 6;             // 0..3 (row within block)
  const int x0 = (t & 63) << 2;     // 4 pixels per thread
  const float* row0 = &tile[(r + 0) * kW];
  const float* row1 = &tile[(r + 1) * kW];
  const float* row2 = &tile[(r + 2) * kW];

#pragma unroll
  for (int j = 0; j < 4; ++j) {
    const int x = x0 + j;
    const int xm = x > 0 ? x - 1 : 0;
    const int xp = x < kW - 1 ? x + 1 : kW - 1;
    const float cv = row1[x];
    float m = row0[xm];
    m = fmaxf(m, row0[x]);  m = fmaxf(m, row0[xp]);
    m = fmaxf(m, row1[xm]); m = fmaxf(m, row1[xp]);
    m = fmaxf(m, row2[xm]); m = fmaxf(m, row2[x]); m = fmaxf(m, row2[xp]);
    if (cv >= m && cv > kThresh) {
      int bin = (int)(cv * (float)kBins);
      bin = bin < 0 ? 0 : (bin > kBins - 1 ? kBins - 1 : bin);
      if (!COLLECT) {
        atomicAdd(&ws[WS_HIST + b * kBins + bin], 1u);
      } else if ((unsigned)bin >= cut) {
        unsigned pos = atomicAdd(&ws[WS_CNT + b], 1u);
        if (pos < (unsigned)kCandCap) {
          ((float*)ws)[WS_CAND_S + b * kCandCap + pos] = cv;
          ws[WS_CAND_I + b * kCandCap + pos] =
              (unsigned)c * kHW + (unsigned)(y0 + r) * kW + (unsigned)x;
        }
      }
    }
  }
}

// ---------------------------------------------------------------------------
// Kernel 2: per-batch cutoff bin = max bin with suffix_count >= TopK (0 if
// fewer than TopK peaks). Two-level 64x64 scan.
// ---------------------------------------------------------------------------
__global__ void cutoff_kernel(unsigned* __restrict__ ws) {
  __shared__ unsigned hbin[kBins];
  __shared__ unsigned gsum[64];
  const int b = blockIdx.x;
  const int t = threadIdx.x;  // 64 threads

  for (int i = t; i < kBins; i += 64) hbin[i] = ws[WS_HIST + b * kBins + i];
  __syncthreads();

  unsigned s = 0;
  for (int j = 0; j < 64; ++j) s += hbin[t * 64 + j];
  gsum[t] = s;
  __syncthreads();

  if (t == 0) {
    unsigned acc = 0;
    int sel = -1;
    for (int g = 63; g >= 0; --g) {
      if (acc + gsum[g] >= (unsigned)kTopK) { sel = g; break; }
      acc += gsum[g];
    }
    unsigned cut = 0;
    if (sel >= 0) {
      unsigned a = acc;
      for (int bin = sel * 64 + 63; bin >= sel * 64; --bin) {
        a += hbin[bin];
        if (a >= (unsigned)kTopK) { cut = (unsigned)bin; break; }
      }
    }
    ws[WS_CUT + b] = cut;
  }
}

// ---------------------------------------------------------------------------
// Kernel 4: per-batch bitonic sort of <=4096 packed keys in LDS, then emit
// ids / scores / bboxes for the top-100 with reference masking semantics
// (masked bbox value = -1 * SCALE = -4).
// Key = (score_bits << 32) | ~index  -> descending score, lowest index first.
// ---------------------------------------------------------------------------
__global__ void finalize_kernel(const unsigned* __restrict__ ws,
                                const float* __restrict__ offset,
                                const float* __restrict__ wh,
                                float* __restrict__ out) {
  __shared__ unsigned long long key[kCandCap];
  const int b = blockIdx.x;
  const unsigned t = threadIdx.x;  // 1024 threads

  unsigned n = ws[WS_CNT + b];
  if (n > (unsigned)kCandCap) n = (unsigned)kCandCap;

  for (unsigned i = t; i < (unsigned)kCandCap; i += 1024) {
    if (i < n) {
      float sv = ((const float*)ws)[WS_CAND_S + b * kCandCap + i];
      unsigned idx = ws[WS_CAND_I + b * kCandCap + i];
      key[i] = ((unsigned long long)__float_as_uint(sv) << 32) |
               (unsigned)(~idx);
    } else {
      key[i] = 0ull;
    }
  }
  __syncthreads();

  // Ascending bitonic sort of 4096 elements.
  for (unsigned k = 2; k <= (unsigned)kCandCap; k <<= 1) {
    for (unsigned j = k >> 1; j > 0; j >>= 1) {
      for (unsigned i = t; i < (unsigned)kCandCap; i += 1024) {
        unsigned ixj = i ^ j;
        if (ixj > i) {
          bool up = ((i & k) == 0);
          unsigned long long a = key[i], c = key[ixj];
          if ((a > c) == up) { key[i] = c; key[ixj] = a; }
        }
      }
      __syncthreads();
    }
  }

  if (t < (unsigned)kTopK) {
    unsigned long long kk = key[kCandCap - 1 - t];  // t-th largest
    float sv = __uint_as_float((unsigned)(kk >> 32));
    float idv, scv, b0, b1, b2, b3;
    if (sv > kThresh) {
      unsigned idx = ~(unsigned)(kk & 0xFFFFFFFFu);
      unsigned ch = idx / (unsigned)kHW;
      unsigned sp = idx % (unsigned)kHW;
      float ys = (float)(sp / (unsigned)kW);
      float xs = (float)(sp % (unsigned)kW);
      const float* offb = offset + (size_t)b * 2 * kHW;
      const float* whb  = wh + (size_t)b * 2 * kHW;
      float ox = offb[sp], oy = offb[kHW + sp];
      float bw = whb[sp],  bh = whb[kHW + sp];
      float cx = xs + ox, cy = ys + oy;
      float hw = bw * 0.5f, hh = bh * 0.5f;
      idv = (float)ch;
      scv = sv;
      b0 = cx - hw; b1 = cy - hh; b2 = cx + hw; b3 = cy + hh;
    } else {
      idv = -1.0f; scv = -1.0f;
      b0 = -1.0f; b1 = -1.0f; b2 = -1.0f; b3 = -1.0f;
    }
    // Output layout: ids[16,100,1] | scores[16,100,1] | bboxes[16,100,4]
    out[b * kTopK + t] = idv;
    out[kB * kTopK + b * kTopK + t] = scv;
    float* bb = out + 2 * kB * kTopK + (size_t)(b * kTopK + t) * 4;
    bb[0] = b0 * kScale; bb[1] = b1 * kScale;
    bb[2] = b2 * kScale; bb[3] = b3 * kScale;
  }
}

// ---------------------------------------------------------------------------
extern "C" void kernel_launch(void* const* d_in, const int* in_sizes, int n_in,
                              void* d_out, int out_size, void* d_ws,
                              size_t ws_size, hipStream_t stream) {
  (void)in_sizes; (void)n_in; (void)out_size; (void)ws_size;
  const float* hm = (const float*)d_in[0];       // [16,80,256,256]
  const float* offset = (const float*)d_in[1];   // [16,2,256,256]
  const float* wh = (const float*)d_in[2];       // [16,2,256,256]
  float* out = (float*)d_out;                    // 9600 floats
  unsigned* ws = (unsigned*)d_ws;                // needs ~787 KB

  init_ws_kernel<<<(WS_ZERO_WORDS + 255) / 256, 256, 0, stream>>>(ws);

  const int nblocks = kB * kC * (kH / 4);        // 81920
  peaks_kernel<false><<<nblocks, 256, 0, stream>>>(hm, ws);
  cutoff_kernel<<<kB, 64, 0, stream>>>(ws);
  peaks_kernel<true><<<nblocks, 256, 0, stream>>>(hm, ws);
  finalize_kernel<<<kB, 1024, 0, stream>>>(ws, offset, wh, out);
}